// CausalAttention_59322088292721
// MI455X (gfx1250) — compile-verified
//
#include <hip/hip_runtime.h>
#include <math.h>

typedef __attribute__((ext_vector_type(2))) float v2f;
typedef __attribute__((ext_vector_type(8))) float v8f;

#define EDIM 1024
#define NTOK 16
#define TPREV 2047
#define TTOT  2048
#define ROWF  4096            // J*H*D floats between consecutive t slots

// ---------------------------------------------------------------------------
// Generic per-token RMS norm:  out[b*sb + j*sj + e] = in[tok*E+e] * 32*g/max(||in_tok||,1e-12)
// grid = 16 (one block per token), block = 256
// ---------------------------------------------------------------------------
__global__ void rmsnorm_k(const float* __restrict__ in, const float* __restrict__ g,
                          float* __restrict__ outb, long sb, long sj) {
    __shared__ float red[256];
    const int tok = blockIdx.x, tid = threadIdx.x;
    const int b = tok >> 2, j = tok & 3;
    const float* ip = in + (long)tok * EDIM;
    float ss = 0.f;
    for (int e = tid; e < EDIM; e += 256) { float v = ip[e]; ss += v * v; }
    red[tid] = ss; __syncthreads();
    for (int s = 128; s > 0; s >>= 1) { if (tid < s) red[tid] += red[tid + s]; __syncthreads(); }
    float nrm = fmaxf(sqrtf(red[0]), 1e-12f);
    const float sc = 32.0f * g[0] / nrm;           // sqrt(1024) = 32
    float* op = outb + (long)b * sb + (long)j * sj;
    for (int e = tid; e < EDIM; e += 256) op[e] = ip[e] * sc;
}

// ---------------------------------------------------------------------------
// Y[16 x 1024] = X[16 x 1024] . W^T + bias  via V_WMMA_F32_16X16X4_F32
// One wave per 16x16 output tile; grid = 64, block = 32 (wave32).
// A-frag: lane(0-15)=row M, lane-half selects K-pair {0,1}/{2,3}.
// B-frag: lane(0-15)=col N (== W row f), same K split.  D: 8 VGPRs.
// ---------------------------------------------------------------------------
__global__ void wmma_gemm_k(const float* __restrict__ X, const float* __restrict__ W,
                            const float* __restrict__ bias, float* __restrict__ Y) {
    const int lane = threadIdx.x & 31;
    const int row  = lane & 15;          // M for A, N for B
    const int kh   = lane >> 4;          // K half (0 or 1)
    const int nb   = blockIdx.x << 4;    // tile column base
    const float* xp = X + row * EDIM + (kh << 1);
    const float* wp = W + (nb + row) * EDIM + (kh << 1);
    v8f acc = {};
#pragma unroll 8
    for (int k0 = 0; k0 < EDIM; k0 += 4) {
        v2f a = *(const v2f*)(xp + k0);
        v2f b = *(const v2f*)(wp + k0);
        acc = __builtin_amdgcn_wmma_f32_16x16x4_f32(
            /*neg_a=*/false, a, /*neg_b=*/false, b,
            /*c_mod=*/(short)0, acc, /*reuse_a=*/false, /*reuse_b=*/false);
    }
    const float bb = bias[nb + row];
#pragma unroll
    for (int r = 0; r < 8; r++) {
        const int m = r + (kh << 3);     // vgpr r: lanes0-15 -> M=r, lanes16-31 -> M=8+r
        Y[m * EDIM + nb + row] = acc[r] + bb;
    }
}

// ---------------------------------------------------------------------------
// Fused: KV-cache concat-copy + softmax attention.
// grid = 256 (one block per (b,j,h)), block = 256.
// Streams kv_k/kv_v (t<2047) input->cache while computing; t=2047 read from
// cache (written there by the k/v rmsnorm kernels).
// ---------------------------------------------------------------------------
__global__ void attn_fused_k(const float* __restrict__ qn, const float* __restrict__ kv_k,
                             const float* __restrict__ kv_v, const float* __restrict__ cvec,
                             float* __restrict__ cache_k, float* __restrict__ cache_v,
                             float* __restrict__ att) {
    __shared__ float qs[64];
    __shared__ float lg[TTOT];
    __shared__ float red[256];
    __shared__ float pv[256];
    const int blk = blockIdx.x;
    const int h   = blk & 15;
    const int tok = blk >> 4;            // b*4 + j
    const int b = tok >> 2, j = tok & 3;
    const int tid = threadIdx.x;
    const long in_b  = (long)b * TPREV * ROWF + j * EDIM + h * 64;  // + t*ROWF
    const long out_b = (long)b * TTOT  * ROWF + j * EDIM + h * 64;
    if (tid < 64) qs[tid] = qn[(long)tok * EDIM + h * 64 + tid];
    __syncthreads();
    const float scale = cvec[tok] * 0.125f;      // c / sqrt(64)

    // ---- pass A: logits, streaming k input -> cache ----
    float lmax = -3.0e38f;
    for (int t = tid; t < TTOT; t += 256) {
        float dot = 0.f;
        if (t < TPREV) {
            const float4* kp = (const float4*)(kv_k + in_b + (long)t * ROWF);
            float4*       cp = (float4*)(cache_k + out_b + (long)t * ROWF);
#pragma unroll
            for (int q4 = 0; q4 < 16; q4++) {
                float4 kk = kp[q4];
                cp[q4] = kk;
                dot += kk.x * qs[4*q4] + kk.y * qs[4*q4+1] + kk.z * qs[4*q4+2] + kk.w * qs[4*q4+3];
            }
        } else {
            const float4* kp = (const float4*)(cache_k + out_b + (long)t * ROWF);
#pragma unroll
            for (int q4 = 0; q4 < 16; q4++) {
                float4 kk = kp[q4];
                dot += kk.x * qs[4*q4] + kk.y * qs[4*q4+1] + kk.z * qs[4*q4+2] + kk.w * qs[4*q4+3];
            }
        }
        const float l = dot * scale;
        lg[t] = l;
        lmax = fmaxf(lmax, l);
    }
    red[tid] = lmax; __syncthreads();
    for (int s = 128; s > 0; s >>= 1) { if (tid < s) red[tid] = fmaxf(red[tid], red[tid + s]); __syncthreads(); }
    const float mx = red[0]; __syncthreads();

    // ---- exp + sum ----
    float lsum = 0.f;
    for (int t = tid; t < TTOT; t += 256) { float e2 = __expf(lg[t] - mx); lg[t] = e2; lsum += e2; }
    red[tid] = lsum; __syncthreads();
    for (int s = 128; s > 0; s >>= 1) { if (tid < s) red[tid] += red[tid + s]; __syncthreads(); }
    const float inv = 1.0f / red[0];

    // ---- pass B: weighted V, streaming v input -> cache ----
    const int d = tid & 63, grp = tid >> 6;      // 4 t-groups x 64 lanes over D
    float accv = 0.f;
    for (int t = grp; t < TTOT; t += 4) {
        float vv;
        if (t < TPREV) {
            vv = kv_v[in_b + (long)t * ROWF + d];
            cache_v[out_b + (long)t * ROWF + d] = vv;
        } else {
            vv = cache_v[out_b + (long)t * ROWF + d];
        }
        accv += lg[t] * vv;
    }
    pv[tid] = accv; __syncthreads();
    if (tid < 64) {
        const float o = (pv[tid] + pv[64 + tid] + pv[128 + tid] + pv[192 + tid]) * inv;
        att[(long)tok * EDIM + h * 64 + tid] = o;
    }
}

// ---------------------------------------------------------------------------
extern "C" void kernel_launch(void* const* d_in, const int* in_sizes, int n_in,
                              void* d_out, int out_size, void* d_ws, size_t ws_size,
                              hipStream_t stream) {
    const float* x      = (const float*)d_in[0];
    const float* c      = (const float*)d_in[1];
    const float* kvk    = (const float*)d_in[2];
    const float* kvv    = (const float*)d_in[3];
    const float* g_norm = (const float*)d_in[4];
    const float* g_q    = (const float*)d_in[5];
    const float* g_k    = (const float*)d_in[6];
    const float* g_v    = (const float*)d_in[7];
    const float* g_out  = (const float*)d_in[8];
    const float* Wq     = (const float*)d_in[9];
    const float* bq     = (const float*)d_in[10];
    const float* Wk     = (const float*)d_in[11];
    const float* bk     = (const float*)d_in[12];
    const float* Wv     = (const float*)d_in[13];
    const float* bv     = (const float*)d_in[14];
    const float* Wo     = (const float*)d_in[15];
    const float* bo     = (const float*)d_in[16];

    float* out     = (float*)d_out;                     // (B,1,J,E) = 16384
    float* cache_k = out + 16384;                       // (B,2048,J,H,D)
    float* cache_v = cache_k + (size_t)TTOT * 4 * ROWF; // +33554432

    float* ws   = (float*)d_ws;                         // 7 x 64KB scratch
    float* xn   = ws;
    float* qraw = ws + 1 * 16384;
    float* kraw = ws + 2 * 16384;
    float* vraw = ws + 3 * 16384;
    float* qn   = ws + 4 * 16384;
    float* att  = ws + 5 * 16384;
    float* oraw = ws + 6 * 16384;

    // 1) x rms-norm
    rmsnorm_k<<<16, 256, 0, stream>>>(x, g_norm, xn, 4096L, 1024L);
    // 2) q,k,v projections (WMMA f32)
    wmma_gemm_k<<<64, 32, 0, stream>>>(xn, Wq, bq, qraw);
    wmma_gemm_k<<<64, 32, 0, stream>>>(xn, Wk, bk, kraw);
    wmma_gemm_k<<<64, 32, 0, stream>>>(xn, Wv, bv, vraw);
    // 3) per-token norms; k,v land directly in cache slot t = 2047
    rmsnorm_k<<<16, 256, 0, stream>>>(qraw, g_q, qn, 4096L, 1024L);
    rmsnorm_k<<<16, 256, 0, stream>>>(kraw, g_k, cache_k + (long)TPREV * ROWF, (long)TTOT * ROWF, 1024L);
    rmsnorm_k<<<16, 256, 0, stream>>>(vraw, g_v, cache_v + (long)TPREV * ROWF, (long)TTOT * ROWF, 1024L);
    // 4) fused cache-concat copy + attention
    attn_fused_k<<<256, 256, 0, stream>>>(qn, kvk, kvv, c, cache_k, cache_v, att);
    // 5) output projection + final norm
    wmma_gemm_k<<<64, 32, 0, stream>>>(att, Wo, bo, oraw);
    rmsnorm_k<<<16, 256, 0, stream>>>(oraw, g_out, out, 4096L, 1024L);
}